// GraphModel_88313117540688
// MI455X (gfx1250) — compile-verified
//
#include <hip/hip_runtime.h>
#include <math.h>

// ---------------------------------------------------------------------------
// GCN on MI455X (gfx1250): fp32 WMMA for dense transforms, atomics for the
// edge scatter (memory-bound part), LDS softmax for the 100-wide blocks.
// ---------------------------------------------------------------------------

typedef __attribute__((ext_vector_type(2))) float v2f;
typedef __attribute__((ext_vector_type(8))) float v8f;

#define HIDDEN 16
#define BLKSZ 100

// ---------------- degree / norm precompute ----------------

__global__ void deg_init_kernel(float* __restrict__ deg, int N) {
    int i = blockIdx.x * blockDim.x + threadIdx.x;
    if (i < N) deg[i] = 1.0f;               // self-loop contributes 1
}

__global__ void deg_edge_kernel(const int* __restrict__ dst, float* __restrict__ deg, int E) {
    int e = blockIdx.x * blockDim.x + threadIdx.x;
    if (e < E) atomicAdd(&deg[dst[e]], 1.0f);
}

__global__ void dinv_kernel(float* __restrict__ dinv, int N) {
    int i = blockIdx.x * blockDim.x + threadIdx.x;
    if (i < N) dinv[i] = rsqrtf(dinv[i]);   // deg >= 1 always (self loops)
}

__global__ void norm_kernel(const int* __restrict__ src, const int* __restrict__ dst,
                            const float* __restrict__ dinv, float* __restrict__ norm, int E) {
    int e = blockIdx.x * blockDim.x + threadIdx.x;
    if (e < E) norm[e] = dinv[src[e]] * dinv[dst[e]];
}

// ---------------- dense transform: out[N,16] = (relu?)in[N,K] @ W[K,16] -----
// One wave computes a 16x16 tile via V_WMMA_F32_16X16X4_F32, K stepped by 4.
// 256 threads = 8 waves = 128 nodes per workgroup. W staged in LDS.

__global__ __launch_bounds__(256)
void gcn_transform_wmma(const float* __restrict__ in, const float* __restrict__ W,
                        float* __restrict__ out, int N, int K, int relu_in) {
    __shared__ float Wlds[128 * HIDDEN];    // max K=128 -> 8 KB
    int tid = threadIdx.x;
    for (int i = tid; i < K * HIDDEN; i += blockDim.x) Wlds[i] = W[i];
    __syncthreads();

    int wave = tid >> 5;
    int lane = tid & 31;
    int half = lane >> 4;                   // 0: K pair {kk,kk+1}, 1: {kk+2,kk+3}
    int l16  = lane & 15;

    long nodeBase = (long)blockIdx.x * 128 + (long)wave * 16;

    v8f c = {};
    for (int kk = 0; kk < K; kk += 4) {
        // A tile: 16x4 fp32, row M = l16, per-lane K pair selected by half
        long node = nodeBase + l16;
        if (node >= N) node = N - 1;        // clamp: keep EXEC all-1 for WMMA
        int k0 = kk + half * 2;
        const float* ip = in + node * (long)K;
        float a0 = ip[k0];
        float a1 = ip[k0 + 1];
        if (relu_in) { a0 = fmaxf(a0, 0.0f); a1 = fmaxf(a1, 0.0f); }
        v2f a = {a0, a1};
        // B tile: 4x16 fp32, N column = l16, K rows mirror A's pairing
        v2f b = { Wlds[k0 * HIDDEN + l16], Wlds[(k0 + 1) * HIDDEN + l16] };
        c = __builtin_amdgcn_wmma_f32_16x16x4_f32(
                /*neg_a=*/false, a, /*neg_b=*/false, b,
                /*c_mod=*/(short)0, c, /*reuse_a=*/false, /*reuse_b=*/false);
    }

    // C/D layout: VGPR r -> M=r (lanes 0-15), M=8+r (lanes 16-31); N = l16
    int col = l16;
    #pragma unroll
    for (int r = 0; r < 8; ++r) {
        long node = nodeBase + half * 8 + r;
        if (node < N) out[node * HIDDEN + col] = c[r];
    }
}

// ---------------- aggregation: agg = b + dinv^2 * z, then += norm*z[src] ----

__global__ void agg_init16_kernel(float* __restrict__ agg, const float* __restrict__ z,
                                  const float* __restrict__ dinv, const float* __restrict__ b,
                                  int N) {
    int i = blockIdx.x * blockDim.x + threadIdx.x;
    if (i >= N * HIDDEN) return;
    int n = i >> 4, f = i & 15;
    float di = dinv[n];
    agg[i] = b[f] + di * di * z[i];
}

__global__ void scatter16_kernel(const int* __restrict__ src, const int* __restrict__ dst,
                                 const float* __restrict__ norm, const float* __restrict__ z,
                                 float* __restrict__ agg, int E) {
    int e = blockIdx.x * blockDim.x + threadIdx.x;
    if (e >= E) return;
    int s = src[e], d = dst[e];
    float w = norm[e];
    const float4* zs = (const float4*)(z + (long)s * HIDDEN);
    float4 v0 = zs[0], v1 = zs[1], v2 = zs[2], v3 = zs[3];
    float* a = agg + (long)d * HIDDEN;
    atomicAdd(a +  0, w * v0.x); atomicAdd(a +  1, w * v0.y);
    atomicAdd(a +  2, w * v0.z); atomicAdd(a +  3, w * v0.w);
    atomicAdd(a +  4, w * v1.x); atomicAdd(a +  5, w * v1.y);
    atomicAdd(a +  6, w * v1.z); atomicAdd(a +  7, w * v1.w);
    atomicAdd(a +  8, w * v2.x); atomicAdd(a +  9, w * v2.y);
    atomicAdd(a + 10, w * v2.z); atomicAdd(a + 11, w * v2.w);
    atomicAdd(a + 12, w * v3.x); atomicAdd(a + 13, w * v3.y);
    atomicAdd(a + 14, w * v3.z); atomicAdd(a + 15, w * v3.w);
}

// ---------------- final layer: zf[n] = relu(agg[n,:]) . Wl ------------------

__global__ void final_dot_kernel(const float* __restrict__ agg, const float* __restrict__ Wl,
                                 float* __restrict__ zf, int N) {
    int n = blockIdx.x * blockDim.x + threadIdx.x;
    if (n >= N) return;
    const float* h = agg + (long)n * HIDDEN;
    float acc = 0.0f;
    #pragma unroll
    for (int f = 0; f < HIDDEN; ++f) acc += fmaxf(h[f], 0.0f) * Wl[f];
    zf[n] = acc;
}

__global__ void aggf_init_kernel(float* __restrict__ aggf, const float* __restrict__ zf,
                                 const float* __restrict__ dinv, const float* __restrict__ bl,
                                 int N) {
    int n = blockIdx.x * blockDim.x + threadIdx.x;
    if (n >= N) return;
    float di = dinv[n];
    aggf[n] = bl[0] + di * di * zf[n];
}

__global__ void scatter1_kernel(const int* __restrict__ src, const int* __restrict__ dst,
                                const float* __restrict__ norm, const float* __restrict__ zf,
                                float* __restrict__ aggf, int E) {
    int e = blockIdx.x * blockDim.x + threadIdx.x;
    if (e >= E) return;
    atomicAdd(&aggf[dst[e]], norm[e] * zf[src[e]]);
}

// ---------------- softmax over blocks of 100 --------------------------------

__global__ __launch_bounds__(128)
void softmax100_kernel(const float* __restrict__ in, float* __restrict__ out) {
    __shared__ float sdata[128];
    int g = blockIdx.x;
    int t = threadIdx.x;
    float v = (t < BLKSZ) ? in[g * BLKSZ + t] : -INFINITY;
    sdata[t] = v;
    __syncthreads();
    for (int s = 64; s > 0; s >>= 1) {
        if (t < s) sdata[t] = fmaxf(sdata[t], sdata[t + s]);
        __syncthreads();
    }
    float mx = sdata[0];
    __syncthreads();
    float ev = (t < BLKSZ) ? expf(v - mx) : 0.0f;
    sdata[t] = ev;
    __syncthreads();
    for (int s = 64; s > 0; s >>= 1) {
        if (t < s) sdata[t] += sdata[t + s];
        __syncthreads();
    }
    float sum = sdata[0];
    if (t < BLKSZ) out[g * BLKSZ + t] = ev / sum;
}

// ---------------------------------------------------------------------------

extern "C" void kernel_launch(void* const* d_in, const int* in_sizes, int n_in,
                              void* d_out, int out_size, void* d_ws, size_t ws_size,
                              hipStream_t stream) {
    const float* x  = (const float*)d_in[0];    // [N,128]
    const int*   ei = (const int*)d_in[1];      // [2,E]
    const float* W0 = (const float*)d_in[2];    // [128,16]
    const float* b0 = (const float*)d_in[3];    // [16]
    const float* Ws = (const float*)d_in[4];    // [5,16,16]
    const float* bs = (const float*)d_in[5];    // [5,16]
    const float* Wl = (const float*)d_in[6];    // [16,1]
    const float* bl = (const float*)d_in[7];    // [1]

    const int INIT_DIM = 128;
    const int N = in_sizes[0] / INIT_DIM;       // 100000
    const int E = in_sizes[1] / 2;              // 3200000
    const int LAYERS = in_sizes[4] / (HIDDEN * HIDDEN);  // 5

    const int* src = ei;
    const int* dst = ei + E;

    // scratch sub-allocation (256-B aligned)
    char* ws = (char*)d_ws;
    size_t off = 0;
    auto alloc = [&](size_t bytes) -> float* {
        float* p = (float*)(ws + off);
        off = (off + bytes + 255) & ~(size_t)255;
        return p;
    };
    float* dinv = alloc((size_t)N * 4);             // degree, then dinv in place
    float* norm = alloc((size_t)E * 4);             // per-edge norm
    float* z    = alloc((size_t)N * HIDDEN * 4);    // transform output
    float* agg  = alloc((size_t)N * HIDDEN * 4);    // aggregation output
    float* zf   = alloc((size_t)N * 4);             // final per-node scalar
    float* aggf = alloc((size_t)N * 4);             // final aggregated scalar
    (void)ws_size;

    const int T = 256;
    int gN   = (N + T - 1) / T;
    int gE   = (E + T - 1) / T;
    int gN16 = (N * HIDDEN + T - 1) / T;
    int gMM  = (N + 127) / 128;                 // 128 nodes per workgroup

    // --- degree / symmetric normalization ---
    deg_init_kernel<<<gN, T, 0, stream>>>(dinv, N);
    deg_edge_kernel<<<gE, T, 0, stream>>>(dst, dinv, E);
    dinv_kernel<<<gN, T, 0, stream>>>(dinv, N);
    norm_kernel<<<gE, T, 0, stream>>>(src, dst, dinv, norm, E);

    // --- layer 0: x[N,128] @ W0 ---
    gcn_transform_wmma<<<gMM, T, 0, stream>>>(x, W0, z, N, INIT_DIM, /*relu_in=*/0);
    agg_init16_kernel<<<gN16, T, 0, stream>>>(agg, z, dinv, b0, N);
    scatter16_kernel<<<gE, T, 0, stream>>>(src, dst, norm, z, agg, E);

    // --- hidden layers: relu fused into the next transform's A-load ---
    for (int i = 0; i < LAYERS; ++i) {
        gcn_transform_wmma<<<gMM, T, 0, stream>>>(agg, Ws + (size_t)i * HIDDEN * HIDDEN,
                                                  z, N, HIDDEN, /*relu_in=*/1);
        agg_init16_kernel<<<gN16, T, 0, stream>>>(agg, z, dinv, bs + (size_t)i * HIDDEN, N);
        scatter16_kernel<<<gE, T, 0, stream>>>(src, dst, norm, z, agg, E);
    }

    // --- final 16 -> 1 layer ---
    final_dot_kernel<<<gN, T, 0, stream>>>(agg, Wl, zf, N);
    aggf_init_kernel<<<gN, T, 0, stream>>>(aggf, zf, dinv, bl, N);
    scatter1_kernel<<<gE, T, 0, stream>>>(src, dst, norm, zf, aggf, E);

    // --- softmax over each block of 100 ---
    softmax100_kernel<<<N / BLKSZ, 128, 0, stream>>>(aggf, (float*)d_out);
}